// TensorProductConvLayer_32366873542777
// MI455X (gfx1250) — compile-verified
//
#include <hip/hip_runtime.h>
#include <math.h>

typedef __attribute__((ext_vector_type(2))) float v2f;
typedef __attribute__((ext_vector_type(8))) float v8f;
typedef __attribute__((ext_vector_type(4))) int   v4i;

#define N_EDGE_F   64
#define N_HID      128
#define N_WNUM     448
#define N_INDIM    320
#define N_OUTDIM   960
#define WAVES_PB   4
#define MTILES_PW  2                       // 2 M-tiles (32 edges) per wave
#define EDGES_PW   (MTILES_PW * 16)
#define EDGES_PB   (WAVES_PB * EDGES_PW)   // 128 edges per block
#define HPAD       (N_HID + 2)             // 130: keeps 8B align of even cols, spreads banks

#define C_ACTNORM    1.41421356237309515f   /* sqrt(2) */
#define C_S1         (C_ACTNORM * 0.125f)   /* sqrt(2)/sqrt(64) */
#define C_S2         0.08838834764831845f   /* 1/sqrt(128) */
#define C_INVSQRT3   0.57735026918962584f
#define C_INVSQRT2   0.70710678118654757f

// ---- CDNA5 async global->LDS staging (guarded; falls back to ld/st) ----
#if defined(__has_builtin)
#if __has_builtin(__builtin_amdgcn_global_load_async_to_lds_b128) && \
    __has_builtin(__builtin_amdgcn_s_wait_asynccnt)
#define USE_ASYNC_LDS 1
#endif
#endif
#ifndef USE_ASYNC_LDS
#define USE_ASYNC_LDS 0
#endif

#if USE_ASYNC_LDS
typedef __attribute__((address_space(1))) v4i* gv4ip;   // global int4*
typedef __attribute__((address_space(3))) v4i* lv4ip;   // LDS int4*
#endif

// out[n, 0:320] = node_attr[n, :]; out[n, 320:960] = 0
__global__ __launch_bounds__(256) void tpconv_init_kernel(
    const float* __restrict__ node_attr, float* __restrict__ out, int n_nodes) {
  int c = blockIdx.x * blockDim.x + threadIdx.x;
  int n = blockIdx.y;
  if (c >= N_OUTDIM || n >= n_nodes) return;
  float v = (c < N_INDIM) ? node_attr[(size_t)n * N_INDIM + c] : 0.0f;
  out[(size_t)n * N_OUTDIM + c] = v;
}

// Tensor-product scatter for one w element (edge row m of a tile, column n).
__device__ __forceinline__ void scatter_tp(
    float wv, int n, float* __restrict__ op, const float* __restrict__ xp,
    float sh0, float s1x, float s1y, float s1z) {
  if (n < 128) {                      // o0: cols [0,128)
    atomicAdd(op + n, wv * xp[n] * sh0);
  } else if (n < 256) {               // o1: cols [192,576), i = n-128
    const int i = n - 128;
    const float t = wv * xp[i];
    atomicAdd(op + 192 + i * 3 + 0, t * s1x);
    atomicAdd(op + 192 + i * 3 + 1, t * s1y);
    atomicAdd(op + 192 + i * 3 + 2, t * s1z);
  } else if (n < 320) {               // o2: cols [576,768), u = n-256
    const int u = n - 256;
    const float t = wv * sh0;
    atomicAdd(op + 576 + u * 3 + 0, t * xp[128 + 3 * u + 0]);
    atomicAdd(op + 576 + u * 3 + 1, t * xp[128 + 3 * u + 1]);
    atomicAdd(op + 576 + u * 3 + 2, t * xp[128 + 3 * u + 2]);
  } else if (n < 384) {               // o3: cols [128,192), u = n-320
    const int u = n - 320;
    const float x = xp[128 + 3 * u], y = xp[128 + 3 * u + 1], z = xp[128 + 3 * u + 2];
    atomicAdd(op + 128 + u, wv * (x * s1x + y * s1y + z * s1z) * C_INVSQRT3);
  } else {                            // o4: cols [768,960), u = n-384
    const int u = n - 384;
    const float x = xp[128 + 3 * u], y = xp[128 + 3 * u + 1], z = xp[128 + 3 * u + 2];
    const float t = wv * C_INVSQRT2;
    atomicAdd(op + 768 + u * 3 + 0, t * (y * s1z - z * s1y));
    atomicAdd(op + 768 + u * 3 + 1, t * (z * s1x - x * s1z));
    atomicAdd(op + 768 + u * 3 + 2, t * (x * s1y - y * s1x));
  }
}

__global__ __launch_bounds__(WAVES_PB * 32) void tpconv_edge_kernel(
    const float* __restrict__ node_attr,   // N x 320
    const int*   __restrict__ edge_index,  // 2 x E
    const float* __restrict__ edge_attr,   // E x 64
    const float* __restrict__ edge_sh,     // E x 4
    const float* __restrict__ fc_w1,       // 64 x 128
    const float* __restrict__ fc_w2,       // 128 x 448
    float* __restrict__ out,               // N x 960
    int E) {
  __shared__ float h_s[WAVES_PB][EDGES_PW][HPAD];   // 65 KB
  __shared__ float b_s[2][N_HID][16];               // 16 KB (double-buffered B tile)
  __shared__ int   src_s[WAVES_PB][EDGES_PW];
  __shared__ int   dst_s[WAVES_PB][EDGES_PW];
  __shared__ float sh_s[WAVES_PB][EDGES_PW][4];

  const int tid  = threadIdx.x;
  const int wave = tid >> 5;
  const int lane = tid & 31;
  const int eb   = blockIdx.x * EDGES_PB + wave * EDGES_PW;

  const int arow = lane & 15;        // M row within tile / N col within tile
  const int kh   = (lane >> 4) * 2;  // K sub-offset: lanes 0-15 -> K0,K1; 16-31 -> K2,K3

  // ---- stage per-edge metadata in LDS (one edge per lane) ----
  {
    int e = eb + lane;
    int ec = (e < E) ? e : (E - 1);
    src_s[wave][lane] = edge_index[ec];
    dst_s[wave][lane] = edge_index[E + ec];
    sh_s[wave][lane][0] = edge_sh[(size_t)ec * 4 + 0];
    sh_s[wave][lane][1] = edge_sh[(size_t)ec * 4 + 1];
    sh_s[wave][lane][2] = edge_sh[(size_t)ec * 4 + 2];
    sh_s[wave][lane][3] = edge_sh[(size_t)ec * 4 + 3];
  }

  // ---- GEMM1: h[32x128] = relu(edge_attr[32x64] @ fc_w1[64x128] / 8) * sqrt(2) ----
  // Keep unrolling modest: fc_w1 is L0-resident, so re-loading fragments is
  // nearly free; hoisting them all would blow past 256 VGPRs (s_set_vgpr_msb
  // thrash).
#pragma unroll 1
  for (int mt = 0; mt < MTILES_PW; ++mt) {
    int er = eb + mt * 16 + arow; if (er >= E) er = E - 1;
    const float* aptr = edge_attr + (size_t)er * N_EDGE_F + kh;
    v2f a1[16];
#pragma unroll
    for (int kk = 0; kk < 16; ++kk) a1[kk] = *(const v2f*)(aptr + kk * 4);

#pragma unroll 2
    for (int nt = 0; nt < N_HID / 16; ++nt) {   // 8 N-tiles
      const int n0 = nt * 16;
      v8f acc = {};
#pragma unroll
      for (int kk = 0; kk < 16; ++kk) {
        const int k0 = kk * 4;
        v2f b;
        b.x = fc_w1[(size_t)(k0 + kh + 0) * N_HID + n0 + arow];
        b.y = fc_w1[(size_t)(k0 + kh + 1) * N_HID + n0 + arow];
        acc = __builtin_amdgcn_wmma_f32_16x16x4_f32(
            false, a1[kk], false, b, (short)0, acc, false, false);
      }
#pragma unroll
      for (int r = 0; r < 8; ++r) {
        const int m = r + (lane >> 4) * 8;
        float v = acc[r];
        v = (v > 0.0f) ? v * C_S1 : 0.0f;
        h_s[wave][mt * 16 + m][n0 + arow] = v;
      }
    }
  }

  // ---- GEMM2 (w = h @ fc_w2 / sqrt(128)) with LDS-staged B + fused scatter ----
  // Cooperative B-tile stage: 128x16 floats, 128 threads x 4 passes x 16B.
  auto stage_btile = [&](int nt, int buf) {
#pragma unroll
    for (int p = 0; p < 4; ++p) {
      const int idx  = p * 128 + tid;       // 0..511
      const int k    = idx >> 2;            // 0..127
      const int part = (idx & 3) * 4;       // 0,4,8,12
      const float* src = fc_w2 + (size_t)k * N_WNUM + nt * 16 + part;
#if USE_ASYNC_LDS
      __builtin_amdgcn_global_load_async_to_lds_b128(
          (gv4ip)src, (lv4ip)&b_s[buf][k][part], 0, 0);
#else
      *(float4*)&b_s[buf][k][part] = *(const float4*)src;
#endif
    }
  };

  stage_btile(0, 0);
  int buf = 0;
#pragma unroll 1
  for (int nt = 0; nt < N_WNUM / 16; ++nt) {  // 28 N-tiles
#if USE_ASYNC_LDS
    __builtin_amdgcn_s_wait_asynccnt(0);
#endif
    __syncthreads();                          // b_s[buf] ready; prev buf free
    if (nt + 1 < N_WNUM / 16) stage_btile(nt + 1, buf ^ 1);

    const int n0 = nt * 16;
    const int n  = n0 + arow;
#pragma unroll
    for (int mt = 0; mt < MTILES_PW; ++mt) {
      v8f acc = {};
#pragma unroll 8
      for (int kk = 0; kk < 32; ++kk) {
        const int k0 = kk * 4;
        const v2f a = *(const v2f*)&h_s[wave][mt * 16 + arow][k0 + kh];
        v2f b;
        b.x = b_s[buf][k0 + kh + 0][arow];
        b.y = b_s[buf][k0 + kh + 1][arow];
        acc = __builtin_amdgcn_wmma_f32_16x16x4_f32(
            false, a, false, b, (short)0, acc, false, false);
      }
      // fused tensor-product scatter (segment boundaries are 16-aligned)
#pragma unroll
      for (int r = 0; r < 8; ++r) {
        const int m  = mt * 16 + r + (lane >> 4) * 8;
        if (eb + m >= E) continue;
        const float wv  = acc[r] * C_S2;
        const int   src = src_s[wave][m];
        const int   dst = dst_s[wave][m];
        scatter_tp(wv, n,
                   out + (size_t)src * N_OUTDIM,
                   node_attr + (size_t)dst * N_INDIM,
                   sh_s[wave][m][0], sh_s[wave][m][1],
                   sh_s[wave][m][2], sh_s[wave][m][3]);
      }
    }
    buf ^= 1;
  }
}

extern "C" void kernel_launch(void* const* d_in, const int* in_sizes, int n_in,
                              void* d_out, int out_size, void* d_ws, size_t ws_size,
                              hipStream_t stream) {
  const float* node_attr  = (const float*)d_in[0];  // N x 320
  const int*   edge_index = (const int*)  d_in[1];  // 2 x E
  const float* edge_attr  = (const float*)d_in[2];  // E x 64
  const float* edge_sh    = (const float*)d_in[3];  // E x 4
  const float* fc_w1      = (const float*)d_in[4];  // 64 x 128
  const float* fc_w2      = (const float*)d_in[5];  // 128 x 448
  float*       out        = (float*)d_out;

  const int n_nodes = in_sizes[0] / N_INDIM;
  const int E       = in_sizes[2] / N_EDGE_F;

  dim3 ig((N_OUTDIM + 255) / 256, n_nodes);
  tpconv_init_kernel<<<ig, 256, 0, stream>>>(node_attr, out, n_nodes);

  const int blocks = (E + EDGES_PB - 1) / EDGES_PB;
  tpconv_edge_kernel<<<blocks, WAVES_PB * 32, 0, stream>>>(
      node_attr, edge_index, edge_attr, edge_sh, fc_w1, fc_w2, out, E);
}